// PyramidLFusion_62818191671962
// MI455X (gfx1250) — compile-verified
//
#include <hip/hip_runtime.h>
#include <cstdint>
#include <cstddef>

// ---------------------------------------------------------------------------
// Types / helpers
// ---------------------------------------------------------------------------
typedef __attribute__((ext_vector_type(2))) float v2f;
typedef __attribute__((ext_vector_type(8))) float v8f;

#define DEV static __device__ __forceinline__

__constant__ float c_bd6[6] = {1.f, 5.f, 10.f, 10.f, 5.f, 1.f}; // blur+down taps (x 1/16 h, 1/64 v)

// Async global->LDS staging (CDNA5). Guarded so compilation never breaks if the
// toolchain spells the builtin differently; fallback is plain loads.
#if defined(__HIP_DEVICE_COMPILE__) && defined(__gfx1250__)
#if __has_builtin(__builtin_amdgcn_global_load_async_to_lds_b32) && \
    __has_builtin(__builtin_amdgcn_s_wait_asynccnt)
#define USE_ASYNC_LDS 1
#endif
#endif

#if USE_ASYNC_LDS
typedef __attribute__((address_space(1))) int as1_int;
typedef __attribute__((address_space(3))) int as3_int;
// Generic->AS1: identical bit pattern. Generic->AS3: low 32 bits of the generic
// LDS VA are the LDS offset, and AS3 pointers are 32-bit, so the integer
// round-trip truncates to exactly the right value (same idiom CK uses for
// __builtin_amdgcn_global_load_lds).
DEV as1_int* to_global(const void* p) { return (as1_int*)(uintptr_t)p; }
DEV as3_int* to_lds(const void* p) { return (as3_int*)(uintptr_t)p; }
#endif

template <bool BOXC, int T>
DEV float coefval(int i) {
  if (BOXC) return ((unsigned)i < (unsigned)T) ? 1.f : 0.f;
  return ((unsigned)i < 6u) ? c_bd6[i] : 0.f;
}

// ---------------------------------------------------------------------------
// Banded-GEMM separable filter on WMMA (one wave = one 16x16 output tile).
//   HORIZ: out[m,n] = sum_t c[t] * in[m, S*N + t - R]   (A=data 16x4, B=coeff 4x16)
//   VERT : out[m,n] = sum_t c[t] * in[S*M + t - R, n]   (A=coeff 16x4, B=data 4x16)
// Window (15*S+T must be mult of 4) staged in LDS (async-to-LDS when available).
// EPI==1: bloom epilogue  dst = max(qpre * exp(-15 * acc*scale), 0.02)
// ---------------------------------------------------------------------------
template <int T, int S, int R, bool HORIZ, bool BOXC, int EPI>
__launch_bounds__(32)
__global__ void filt_wmma(const float* __restrict__ src, float* __restrict__ dst,
                          const float* __restrict__ epi_in,
                          int inH, int inW, int outH, int outW,
                          int tilesX, int tilesY, float scale) {
  constexpr int WIN = 15 * S + T;   // 76 (box) or 36 (blur-down)
  constexpr int G = WIN / 4;
  __shared__ float win[16 * WIN];

  const int lane = threadIdx.x;
  const int tilesPerImg = tilesX * tilesY;
  const int img = blockIdx.x / tilesPerImg;
  const int t = blockIdx.x - img * tilesPerImg;
  const int ty = t / tilesX;
  const int tx = t - ty * tilesX;
  const int rowO = ty * 16, colO = tx * 16;
  const float* s = src + (size_t)img * inH * inW;

  if (HORIZ) {
    const int c0 = S * colO - R;
    for (int idx = lane; idx < 16 * WIN; idx += 32) {  // 16*WIN % 32 == 0
      const int i = idx / WIN, j = idx - i * WIN;
      const int r = rowO + i;
      const int c = c0 + j;
      const bool ok = (c >= 0) & (c < inW);
#if USE_ASYNC_LDS
      if (ok)
        __builtin_amdgcn_global_load_async_to_lds_b32(
            to_global(s + (size_t)r * inW + c), to_lds(&win[i * WIN + j]), 0, 0);
      else
        win[i * WIN + j] = 0.f;
#else
      win[i * WIN + j] = ok ? s[(size_t)r * inW + c] : 0.f;
#endif
    }
  } else {
    const int r0 = S * rowO - R;
    for (int idx = lane; idx < 16 * WIN; idx += 32) {
      const int j = idx >> 4, n = idx & 15;
      const int r = r0 + j;
      const bool ok = (r >= 0) & (r < inH);
#if USE_ASYNC_LDS
      if (ok)
        __builtin_amdgcn_global_load_async_to_lds_b32(
            to_global(s + (size_t)r * inW + colO + n), to_lds(&win[j * 16 + n]), 0, 0);
      else
        win[j * 16 + n] = 0.f;
#else
      win[j * 16 + n] = ok ? s[(size_t)r * inW + colO + n] : 0.f;
#endif
    }
  }
#if USE_ASYNC_LDS
  __builtin_amdgcn_s_wait_asynccnt(0);
#endif
  __syncthreads();

  // Lane roles (ISA 16x4 f32 A layout / mirrored 4x16 B layout):
  //   m/n = lane&15 ; k pair = {2*(lane>>4), 2*(lane>>4)+1}
  const int hf = lane >> 4;
  const int mn = lane & 15;
  v8f acc = {0.f, 0.f, 0.f, 0.f, 0.f, 0.f, 0.f, 0.f};
#pragma unroll
  for (int g = 0; g < G; ++g) {
    const int k0 = 4 * g + 2 * hf;
    v2f a, b;
    if (HORIZ) {
      a.x = win[mn * WIN + k0];
      a.y = win[mn * WIN + k0 + 1];
      const int i0 = k0 - S * mn;
      b.x = coefval<BOXC, T>(i0);
      b.y = coefval<BOXC, T>(i0 + 1);
    } else {
      const int i0 = k0 - S * mn;
      a.x = coefval<BOXC, T>(i0);
      a.y = coefval<BOXC, T>(i0 + 1);
      b.x = win[k0 * 16 + mn];
      b.y = win[(k0 + 1) * 16 + mn];
    }
    acc = __builtin_amdgcn_wmma_f32_16x16x4_f32(false, a, false, b, (short)0, acc,
                                                false, false);
  }

  // C/D layout: lane -> N = lane&15 ; VGPR v -> M = v + 8*(lane>>4)
  const size_t ob = (size_t)img * outH * outW;
#pragma unroll
  for (int v = 0; v < 8; ++v) {
    const int m = v + 8 * hf;
    const size_t o = ob + (size_t)(rowO + m) * outW + (colO + mn);
    float val = acc[v] * scale;
    if (EPI == 1) {
      const float bloom = __expf(-15.0f * val);
      dst[o] = fmaxf(epi_in[o] * bloom, 0.02f);
    } else {
      dst[o] = val;
    }
  }
}

// ---------------------------------------------------------------------------
// Pointwise / reduction kernels
// ---------------------------------------------------------------------------
__global__ void k_pre(const float* __restrict__ L, float* __restrict__ qpre,
                      float* __restrict__ blown, size_t n) {
  size_t i = (size_t)blockIdx.x * blockDim.x + threadIdx.x;
  if (i >= n) return;
  float x = L[i];
  float t = 2.f * x - 1.f;
  float bq = 1.f - t * t; bq = bq * bq;
  float sat  = 1.f / (1.f + __expf(15.f * (x - 0.88f)));   // sigmoid(-15(x-.88))
  float dark = 1.f / (1.f + __expf(-25.f * (x - 0.04f)));  // sigmoid(25(x-.04))
  qpre[i]  = bq * sat * dark;
  blown[i] = fmaxf(x - 0.88f, 0.f);
}

__global__ void k_red1(const float* __restrict__ x, float* __restrict__ part) {
  __shared__ float sm[256];
  const int b = blockIdx.x;
  const size_t base = (size_t)b * 16384;
  float s = 0.f;
  for (int k = threadIdx.x; k < 16384; k += 256) s += x[base + k];
  sm[threadIdx.x] = s;
  __syncthreads();
  for (int st = 128; st > 0; st >>= 1) {
    if ((int)threadIdx.x < st) sm[threadIdx.x] += sm[threadIdx.x + st];
    __syncthreads();
  }
  if (threadIdx.x == 0) part[b] = sm[0];
}

__global__ void k_red2(const float* __restrict__ part, float* __restrict__ mean,
                       float invLen, int clampLow) {
  __shared__ float sm[64];
  const int img = blockIdx.x;
  sm[threadIdx.x] = part[img * 64 + threadIdx.x];
  __syncthreads();
  for (int st = 32; st > 0; st >>= 1) {
    if ((int)threadIdx.x < st) sm[threadIdx.x] += sm[threadIdx.x + st];
    __syncthreads();
  }
  if (threadIdx.x == 0) {
    float m = sm[0] * invLen;
    if (clampLow) m = fmaxf(m, 0.05f);
    mean[img] = m;
  }
}

__global__ void k_weights(const float* __restrict__ q, float* __restrict__ w,
                          int HW, int Bn, int Nn) {
  size_t i = (size_t)blockIdx.x * blockDim.x + threadIdx.x;
  if (i >= (size_t)Bn * HW) return;
  int b = (int)(i / HW), p = (int)(i - (size_t)b * HW);
  size_t base = (size_t)b * Nn * HW + p;
  float v[8];
  float s = 0.f;
  for (int n = 0; n < Nn; ++n) { v[n] = q[base + (size_t)n * HW]; s += v[n]; }
  float inv = 1.f / (s + 1e-6f);
  for (int n = 0; n < Nn; ++n) w[base + (size_t)n * HW] = v[n] * inv;
}

__global__ void k_target(const float* __restrict__ Lm, const float* __restrict__ wm,
                         float* __restrict__ tgt, int Bn, int Nn) {
  int b = threadIdx.x;
  if (b < Bn) {
    float num = 0.f, den = 0.f;
    for (int n = 0; n < Nn; ++n) {
      num += Lm[b * Nn + n] * wm[b * Nn + n];
      den += wm[b * Nn + n];
    }
    tgt[b] = num / (den + 1e-6f);
  }
}

__global__ void k_comp(const float* __restrict__ L, const float* __restrict__ Lm,
                       const float* __restrict__ tgt, float* __restrict__ outp,
                       size_t n, int HW, int Nn) {
  size_t i = (size_t)blockIdx.x * blockDim.x + threadIdx.x;
  if (i >= n) return;
  int img = (int)(i / HW);
  int b = img / Nn;
  float v = L[i] * (tgt[b] / Lm[img]);
  outp[i] = fminf(fmaxf(v, 0.f), 1.f);
}

// bilinear 2x upsample weights, half-pixel centers, edge-clamped
DEV void upc(int o, int n, int& i0, int& i1, float& w0, float& w1) {
  int m = o >> 1;
  if (o & 1) { i0 = m; i1 = (m + 1 < n) ? m + 1 : n - 1; w0 = 0.75f; w1 = 0.25f; }
  else       { i0 = (m > 0) ? m - 1 : 0; i1 = m;         w0 = 0.25f; w1 = 0.75f; }
}

DEV float up_sample(const float* s, int h, int w, int y, int x) {
  int y0, y1, x0, x1; float wy0, wy1, wx0, wx1;
  upc(y, h, y0, y1, wy0, wy1);
  upc(x, w, x0, x1, wx0, wx1);
  return wy0 * (wx0 * s[(size_t)y0 * w + x0] + wx1 * s[(size_t)y0 * w + x1]) +
         wy1 * (wx0 * s[(size_t)y1 * w + x0] + wx1 * s[(size_t)y1 * w + x1]);
}

__global__ void k_up2_sub(float* __restrict__ hi, const float* __restrict__ lo,
                          int h, int w, int nimg) {
  size_t HW2 = (size_t)4 * h * w;
  size_t i = (size_t)blockIdx.x * blockDim.x + threadIdx.x;
  if (i >= (size_t)nimg * HW2) return;
  int img = (int)(i / HW2);
  int p = (int)(i - (size_t)img * HW2);
  int W2 = 2 * w;
  int y = p / W2, x = p - y * W2;
  hi[i] -= up_sample(lo + (size_t)img * h * w, h, w, y, x);
}

__global__ void k_up2_add(const float* __restrict__ lo, const float* __restrict__ add,
                          float* __restrict__ dst, int h, int w, int nimg, int clip) {
  size_t HW2 = (size_t)4 * h * w;
  size_t i = (size_t)blockIdx.x * blockDim.x + threadIdx.x;
  if (i >= (size_t)nimg * HW2) return;
  int img = (int)(i / HW2);
  int p = (int)(i - (size_t)img * HW2);
  int W2 = 2 * w;
  int y = p / W2, x = p - y * W2;
  float v = up_sample(lo + (size_t)img * h * w, h, w, y, x) + add[i];
  if (clip) v = fminf(fmaxf(v, 0.f), 1.f);
  dst[i] = v;
}

__global__ void k_fuse(const float* __restrict__ Ll, const float* __restrict__ wl,
                       float* __restrict__ outp, int HWl, int Bn, int Nn) {
  size_t i = (size_t)blockIdx.x * blockDim.x + threadIdx.x;
  if (i >= (size_t)Bn * HWl) return;
  int b = (int)(i / HWl), p = (int)(i - (size_t)b * HWl);
  size_t base = (size_t)b * Nn * HWl + p;
  float s = 1e-6f, acc = 0.f;
  for (int n = 0; n < Nn; ++n) {
    float ww = wl[base + (size_t)n * HWl];
    s += ww;
    acc += ww * Ll[base + (size_t)n * HWl];
  }
  outp[i] = acc / s;
}

// ---------------------------------------------------------------------------
// Host orchestration
// ---------------------------------------------------------------------------
extern "C" void kernel_launch(void* const* d_in, const int* in_sizes, int n_in,
                              void* d_out, int out_size, void* d_ws, size_t ws_size,
                              hipStream_t stream) {
  (void)in_sizes; (void)n_in; (void)out_size; (void)ws_size;
  const float* L = (const float*)d_in[0];
  float* out = (float*)d_out;

  constexpr int B = 4, N = 5, H = 1024, W = 1024;
  constexpr int NIMG = B * N;
  constexpr size_t HW = (size_t)H * W;
  constexpr size_t NP = (size_t)NIMG * HW;  // 20,971,520 elems

  float* ws = (float*)d_ws;
  float* bufA = ws;            // blown -> blur-down tmp -> fused + collapse ping/pong
  float* bufB = bufA + NP;     // box tmpH -> L_comp (gauss-L level0 / lap0, in place)
  float* qbuf = bufB + NP;     // qpre -> quality -> pyrL | pyrW (quality dead by then)
  float* sm_area = qbuf + NP;
  float* partials = sm_area;        // 1280
  float* L_mean = sm_area + 1280;   // 20
  float* w_mean = sm_area + 1300;   // 20
  float* target = sm_area + 1320;   // 4

  float* fused_out = out;           // (4,1,1024,1024)
  float* w_out = out + B * HW;      // (4,5,1,1024,1024)

  const int hs[5] = {1024, 512, 256, 128, 64};

  // 1) pointwise quality-precursor + blown
  k_pre<<<(unsigned)(NP / 256), 256, 0, stream>>>(L, qbuf, bufA, NP);

  // 2) per-image mean of L (deterministic two-stage)
  k_red1<<<NIMG * 64, 256, 0, stream>>>(L, partials);
  k_red2<<<NIMG, 64, 0, stream>>>(partials, L_mean, 1.f / (float)HW, 1);

  // 3) 61-tap separable box on blown, WMMA banded-GEMM; vertical pass fuses
  //    bloom epilogue: quality = max(qpre * exp(-15*box/3721), 0.02)  (in place)
  filt_wmma<61, 1, 30, true,  true, 0><<<NIMG * 64 * 64, 32, 0, stream>>>(
      bufA, bufB, nullptr, H, W, H, W, 64, 64, 1.0f);
  filt_wmma<61, 1, 30, false, true, 1><<<NIMG * 64 * 64, 32, 0, stream>>>(
      bufB, qbuf, qbuf, H, W, H, W, 64, 64, 1.0f / 3721.0f);

  // 4) weights -> output w; per-image mean of w; target_L
  k_weights<<<(unsigned)((B * HW) / 256), 256, 0, stream>>>(qbuf, w_out, (int)HW, B, N);
  k_red1<<<NIMG * 64, 256, 0, stream>>>(w_out, partials);
  k_red2<<<NIMG, 64, 0, stream>>>(partials, w_mean, 1.f / (float)HW, 0);
  k_target<<<1, 32, 0, stream>>>(L_mean, w_mean, target, B, N);

  // 5) exposure-compensated L -> bufB (level 0 of L pyramid)
  k_comp<<<(unsigned)(NP / 256), 256, 0, stream>>>(L, L_mean, target, bufB, NP, (int)HW, N);

  // pyramid level pointers (levels 1..4 live in qbuf; quality is dead now)
  const size_t PYR = (size_t)NIMG * (512 * 512 + 256 * 256 + 128 * 128 + 64 * 64);
  float* pyrL = qbuf;
  float* pyrW = qbuf + PYR;
  float* pl[5]; float* pw[5];
  pl[0] = bufB; pw[0] = w_out;
  {
    size_t po = 0;
    for (int l = 1; l < 5; ++l) {
      pl[l] = pyrL + po; pw[l] = pyrW + po;
      po += (size_t)NIMG * hs[l] * hs[l];
    }
  }

  // 6) Gaussian pyramids of L_comp and w: blur(5x5)+avgpool2 as 6-tap stride-2
  //    separable WMMA filters (h: /16, v: /64). bufA is the half-width tmp.
  for (int l = 0; l < 4; ++l) {
    const int h = hs[l], w2 = hs[l] / 2, ho = hs[l + 1];
    const int txh = w2 / 16, tyh = h / 16;
    const int txv = w2 / 16, tyv = ho / 16;
    filt_wmma<6, 2, 2, true,  false, 0><<<NIMG * txh * tyh, 32, 0, stream>>>(
        pl[l], bufA, nullptr, h, h, h, w2, txh, tyh, 1.f / 16.f);
    filt_wmma<6, 2, 2, false, false, 0><<<NIMG * txv * tyv, 32, 0, stream>>>(
        bufA, pl[l + 1], nullptr, h, w2, ho, w2, txv, tyv, 1.f / 64.f);
    filt_wmma<6, 2, 2, true,  false, 0><<<NIMG * txh * tyh, 32, 0, stream>>>(
        pw[l], bufA, nullptr, h, h, h, w2, txh, tyh, 1.f / 16.f);
    filt_wmma<6, 2, 2, false, false, 0><<<NIMG * txv * tyv, 32, 0, stream>>>(
        bufA, pw[l + 1], nullptr, h, w2, ho, w2, txv, tyv, 1.f / 64.f);
  }

  // 7) Laplacian of L pyramid, in place (ascending levels)
  for (int l = 0; l < 4; ++l) {
    const int hl = hs[l + 1];
    const size_t n = (size_t)NIMG * 4 * hl * hl;
    k_up2_sub<<<(unsigned)(n / 256), 256, 0, stream>>>(pl[l], pl[l + 1], hl, hl, NIMG);
  }

  // 8) per-level fused = sum_n L_l * w_l / (sum_n w_l + 1e-6)  -> bufA
  size_t f_off[5];
  {
    size_t po = 0;
    for (int l = 0; l < 5; ++l) { f_off[l] = po; po += (size_t)B * hs[l] * hs[l]; }
  }
  for (int l = 0; l < 5; ++l) {
    const int hl = hs[l];
    const size_t n = (size_t)B * hl * hl;
    k_fuse<<<(unsigned)(n / 256), 256, 0, stream>>>(pl[l], pw[l], bufA + f_off[l],
                                                    hl * hl, B, N);
  }

  // 9) collapse (ping/pong in bufA after the fused levels), clip on final level
  float* ping = bufA + (f_off[4] + (size_t)B * 64 * 64);
  float* pong = ping + (size_t)B * 512 * 512;
  float* cur = bufA + f_off[4];
  for (int l = 3; l >= 0; --l) {
    const int hl = hs[l + 1];
    float* dstp = (l == 0) ? fused_out : ((l & 1) ? ping : pong);
    const size_t n = (size_t)B * 4 * hl * hl;
    k_up2_add<<<(unsigned)(n / 256), 256, 0, stream>>>(cur, bufA + f_off[l], dstp,
                                                       hl, hl, B, l == 0 ? 1 : 0);
    cur = dstp;
  }
}